// Dens_compute_81999515615358
// MI455X (gfx1250) — compile-verified
//
#include <hip/hip_runtime.h>

// CDNA5 (gfx1250) wave32 WMMA types
typedef __attribute__((ext_vector_type(2))) float v2f;   // 16x4 / 4x16 f32 operand: 2 VGPRs/lane
typedef __attribute__((ext_vector_type(4))) float v4f;
typedef __attribute__((ext_vector_type(8))) float v8f;   // 16x16 f32 accumulator: 8 VGPRs/lane

#define D_THRESH_SQ 0.25f
#define BIG_NORM    3.0e38f

__global__ void zero_counts_kernel(int* __restrict__ out, int n) {
    int i = blockIdx.x * blockDim.x + threadIdx.x;
    if (i < n) out[i] = 0;
}

// pad4[i] = {x, y, z, ||p||^2} for i < M, else {0,0,0,BIG} (never counted).
// Padding to a multiple of 16 rows removes all tail handling from the hot loop.
__global__ void prep_a_kernel(const float* __restrict__ pad, v4f* __restrict__ pad4,
                              int M, int Mpad) {
    int i = blockIdx.x * blockDim.x + threadIdx.x;
    if (i >= Mpad) return;
    v4f r;
    if (i < M) {
        float x = pad[i * 3 + 0], y = pad[i * 3 + 1], z = pad[i * 3 + 2];
        r[0] = x; r[1] = y; r[2] = z; r[3] = fmaf(x, x, fmaf(y, y, z * z));
    } else {
        r[0] = 0.f; r[1] = 0.f; r[2] = 0.f; r[3] = BIG_NORM;
    }
    pad4[i] = r;
}

__device__ __forceinline__ v8f wmma_tile(v2f a, v2f b) {
    v8f acc = {};
    // 8 args: (neg_a, A, neg_b, B, c_mod, C, reuse_a, reuse_b)
    return __builtin_amdgcn_wmma_f32_16x16x4_f32(false, a, false, b,
                                                 (short)0, acc, false, false);
}

__device__ __forceinline__ int count8(v8f acc, float tn) {
    int c = 0;
    #pragma unroll
    for (int j = 0; j < 8; ++j) c += (acc[j] <= tn) ? 1 : 0;
    return c;
}

// Build the loop-invariant B operand + per-lane threshold.
// f32 4x16 B layout: lanes 0-15: v0=K0,v1=K1; lanes 16-31: v0=K2,v1=K3 (N = lane&15)
__device__ __forceinline__ void make_b(const float* __restrict__ pc, int n, int N,
                                       bool hi, v2f& bmat, float& tn) {
    const int nc = (n < N) ? n : 0;
    const float bx = pc[nc * 3 + 0];
    const float by = pc[nc * 3 + 1];
    const float bz = pc[nc * 3 + 2];
    bmat[0] = hi ? (-2.0f * bz) : (-2.0f * bx);
    bmat[1] = hi ? 1.0f : (-2.0f * by);
    tn = (n < N) ? (D_THRESH_SQ - fmaf(bx, bx, fmaf(by, by, bz * bz))) : -3.0e38f;
}

// Fast path: A operands pre-packed as [Mpad][4] = {x,y,z,nrm}. Each lane's
// operand pair is a single aligned b64 load: no VALU prep, no divergence.
__global__ __launch_bounds__(256)
void density_count_wmma_fast(const float* __restrict__ pc,    // [N,3]
                             const float* __restrict__ pad4,  // [Mpad][4]
                             int* __restrict__ out, int N, int mtiles) {
    const int lane = threadIdx.x & 31;
    const int wave = threadIdx.x >> 5;
    const int row  = lane & 15;
    const bool hi  = (lane >> 4) != 0;   // lanes 16-31 take {z, nrm}; 0-15 take {x, y}

    const int n = blockIdx.x * 16 + row;
    v2f bmat; float tn;
    make_b(pc, n, N, hi, bmat, tn);

    const int stride = gridDim.y * (blockDim.x >> 5);
    int t = blockIdx.y * (blockDim.x >> 5) + wave;
    // float index of this lane's 8-byte A chunk for tile t
    const float* ap = pad4 + ((t * 16 + row) * 4 + (hi ? 2 : 0));
    const int step = stride * 64;        // floats per stride of 16 rows * 4

    int cnt = 0;
    // unroll x2: both b64 loads issue before the first WMMA's wait
    for (; t + stride < mtiles; t += 2 * stride) {
        v2f a0 = *(const v2f*)(ap);
        v2f a1 = *(const v2f*)(ap + step);
        ap += 2 * step;
        cnt += count8(wmma_tile(a0, bmat), tn);
        cnt += count8(wmma_tile(a1, bmat), tn);
    }
    if (t < mtiles) {
        v2f a0 = *(const v2f*)(ap);
        cnt += count8(wmma_tile(a0, bmat), tn);
    }

    cnt += __shfl_xor(cnt, 16, 32);      // rows 0-7 half + rows 8-15 half, same n
    if (!hi && n < N) atomicAdd(&out[n], cnt);
}

// Exact bitwise lane-select: (m & a) | (~m & b) -> v_bfi_b32; cannot be
// re-branched into divergent control flow by the compiler.
__device__ __forceinline__ float sel_bits(int m, float a, float b) {
    return __int_as_float((m & __float_as_int(a)) | (~m & __float_as_int(b)));
}

// Fallback path (workspace too small): in-loop operand build, branch-free.
__global__ __launch_bounds__(256)
void density_count_wmma_blend(const float* __restrict__ pc,   // [N,3]
                              const float* __restrict__ pad,  // [M,3]
                              int* __restrict__ out, int N, int M) {
    const int lane = threadIdx.x & 31;
    const int wave = threadIdx.x >> 5;
    const int row  = lane & 15;
    const bool hi  = (lane >> 4) != 0;
    const int msk  = hi ? -1 : 0;        // all-ones on hi lanes

    const int n = blockIdx.x * 16 + row;
    v2f bmat; float tn;
    make_b(pc, n, N, hi, bmat, tn);

    const int stride    = gridDim.y * (blockDim.x >> 5);
    const int fullTiles = M >> 4;
    int t = blockIdx.y * (blockDim.x >> 5) + wave;
    int cnt = 0;

    for (; t < fullTiles; t += stride) {
        const int m = t * 16 + row;
        const float ax = pad[m * 3 + 0];
        const float ay = pad[m * 3 + 1];
        const float az = pad[m * 3 + 2];
        const float nrm = fmaf(ax, ax, fmaf(ay, ay, az * az));
        v2f amat;
        amat[0] = sel_bits(msk, az, ax);
        amat[1] = sel_bits(msk, nrm, ay);
        cnt += count8(wmma_tile(amat, bmat), tn);
    }
    if ((M & 15) && t == fullTiles) {    // ragged tail tile, one owner wave
        const int m  = t * 16 + row;
        const int mc = (m < M) ? m : 0;
        const float ax = pad[mc * 3 + 0];
        const float ay = pad[mc * 3 + 1];
        const float az = pad[mc * 3 + 2];
        const float nrm = fmaf(ax, ax, fmaf(ay, ay, az * az));
        const int okm = (m < M) ? -1 : 0;
        v2f amat;
        amat[0] = sel_bits(okm, sel_bits(msk, az, ax), 0.0f);
        amat[1] = sel_bits(okm, sel_bits(msk, nrm, ay), sel_bits(msk, BIG_NORM, 0.0f));
        cnt += count8(wmma_tile(amat, bmat), tn);
    }

    cnt += __shfl_xor(cnt, 16, 32);
    if (!hi && n < N) atomicAdd(&out[n], cnt);
}

extern "C" void kernel_launch(void* const* d_in, const int* in_sizes, int n_in,
                              void* d_out, int out_size, void* d_ws, size_t ws_size,
                              hipStream_t stream) {
    const float* pc  = (const float*)d_in[0];  // pointcloud        [N,3]
    const float* pad = (const float*)d_in[1];  // pointcloud_padding[M,3]
    int* out = (int*)d_out;                    // int32 counts [N]

    const int N = in_sizes[0] / 3;
    const int M = in_sizes[1] / 3;
    const int mtiles = (M + 15) >> 4;
    const int Mpad   = mtiles * 16;
    const size_t wsNeed = (size_t)Mpad * 4 * sizeof(float);

    zero_counts_kernel<<<(out_size + 255) / 256, 256, 0, stream>>>(out, out_size);

    dim3 grid((N + 15) / 16, 4, 1);   // 1250 n-strips x 32 waves striding over m-tiles
    if (ws_size >= wsNeed) {
        prep_a_kernel<<<(Mpad + 255) / 256, 256, 0, stream>>>(pad, (v4f*)d_ws, M, Mpad);
        density_count_wmma_fast<<<grid, 256, 0, stream>>>(
            pc, (const float*)d_ws, out, N, mtiles);
    } else {
        density_count_wmma_blend<<<grid, 256, 0, stream>>>(pc, pad, out, N, M);
    }
}